// AttentionMLP_Autograd_89584427860660
// MI455X (gfx1250) — compile-verified
//
#include <hip/hip_runtime.h>

// Problem sizes from the reference
#define Bn 4
#define In 512
#define Jn 512
#define Cn 512
#define IT 8   // i-rows per thread in the sigmoid kernel

typedef float v2f __attribute__((ext_vector_type(2)));
typedef float v8f __attribute__((ext_vector_type(8)));

// ---------------------------------------------------------------------------
// Kernel A: attention_logits[b,i,j] = sum_c Q[b,i,c]*K[b,j,c]
// One wave32 computes one 16x16 output tile via V_WMMA_F32_16X16X4_F32,
// k-loop over C in steps of 4. (Lowering confirmed: v_wmma_f32_16x16x4_f32
// pipelined against global_load_b64 clauses.)
// ---------------------------------------------------------------------------
__global__ __launch_bounds__(32) void attn_logits_wmma(
    const float* __restrict__ Q, const float* __restrict__ K,
    float* __restrict__ attn)
{
    const int lane = threadIdx.x;        // 0..31, EXEC all ones (WMMA requirement)
    const int row  = lane & 15;
    const int grp  = lane >> 4;          // 0 or 1

    const int b  = blockIdx.z;
    const int i0 = blockIdx.y << 4;
    const int j0 = blockIdx.x << 4;

    const float* qrow = Q + ((size_t)b * In + (size_t)(i0 + row)) * Cn + 2 * grp;
    const float* krow = K + ((size_t)b * Jn + (size_t)(j0 + row)) * Cn + 2 * grp;

    v8f acc = {};
#pragma unroll 8
    for (int k0 = 0; k0 < Cn; k0 += 4) {
        v2f a  = *(const v2f*)(qrow + k0);
        v2f bm = *(const v2f*)(krow + k0);
        // (neg_a, A, neg_b, B, c_mod, C, reuse_a, reuse_b)
        acc = __builtin_amdgcn_wmma_f32_16x16x4_f32(
            false, a, false, bm, (short)0, acc, false, false);
    }

    // D[m,n] with m = v + 8*grp, n = row
    float* out = attn + ((size_t)b * In + (size_t)(i0 + 8 * grp)) * Jn
                      + (size_t)(j0 + row);
#pragma unroll
    for (int v = 0; v < 8; ++v)
        out[(size_t)v * Jn] = acc[v];
}

// ---------------------------------------------------------------------------
// Sigmoid via a single transcendental where the HW supports it:
//   sigmoid(x) = 0.5 + 0.5*tanh(0.5*x)   -> one v_tanh_f32 (gfx1250 TRANS op)
// Fallback: rcp(1 + exp2(-x*log2e))      -> v_exp_f32 + v_rcp_f32
// ---------------------------------------------------------------------------
#if __has_builtin(__builtin_amdgcn_tanhf)
  #define HAVE_HW_TANH 1
  __device__ __forceinline__ float hw_tanh(float x) { return __builtin_amdgcn_tanhf(x); }
#elif __has_builtin(__builtin_amdgcn_tanh_f32)
  #define HAVE_HW_TANH 1
  __device__ __forceinline__ float hw_tanh(float x) { return __builtin_amdgcn_tanh_f32(x); }
#else
  #define HAVE_HW_TANH 0
#endif

// ---------------------------------------------------------------------------
// Kernel B: output[b,i,c] = mean_j sigmoid(Q[b,i,c]*K[b,j,c] + bias[c])
// Thread owns one channel c (coalesced) and IT consecutive i-rows.
// Inner body (tanh path): fma -> v_tanh_f32 -> add  (2 VALU + 1 trans/elem),
// the fma/add co-execute under the quarter-rate trans pipe.
// ---------------------------------------------------------------------------
__global__ __launch_bounds__(256) void sigmoid_mean(
    const float* __restrict__ Q, const float* __restrict__ K,
    const float* __restrict__ bias, float* __restrict__ out)
{
    const int c  = (blockIdx.x << 8) + threadIdx.x;   // 0..Cn-1
    const int i0 = blockIdx.y * IT;
    const int b  = blockIdx.z;

#if HAVE_HW_TANH
    // t = 0.5*(q*k + bias) ; sigma = 0.5 + 0.5*tanh(t)
    const float bh = bias[c] * 0.5f;
    float qh[IT], acc[IT];
#pragma unroll
    for (int r = 0; r < IT; ++r) {
        qh[r]  = Q[((size_t)b * In + (size_t)(i0 + r)) * Cn + c] * 0.5f;
        acc[r] = 0.0f;
    }

    const float* kp = K + (size_t)b * Jn * Cn + c;
#pragma unroll 4
    for (int j = 0; j < Jn; ++j) {
        const float kv = kp[(size_t)j * Cn];
#pragma unroll
        for (int r = 0; r < IT; ++r) {
            float t = __builtin_fmaf(qh[r], kv, bh);
            acc[r] += hw_tanh(t);                      // v_tanh_f32
        }
    }

#pragma unroll
    for (int r = 0; r < IT; ++r)
        out[((size_t)b * In + (size_t)(i0 + r)) * Cn + c] =
            __builtin_fmaf(acc[r], 0.5f / Jn, 0.5f);   // 0.5 + (0.5/J)*sum(tanh)
#else
    // Fallback: sigma = rcp(1 + exp2(-(q*k+bias)*log2e))
    const float NEG_LOG2E = -1.4426950408889634f;
    const float bs = bias[c] * NEG_LOG2E;
    float qs[IT], acc[IT];
#pragma unroll
    for (int r = 0; r < IT; ++r) {
        qs[r]  = Q[((size_t)b * In + (size_t)(i0 + r)) * Cn + c] * NEG_LOG2E;
        acc[r] = 0.0f;
    }

    const float* kp = K + (size_t)b * Jn * Cn + c;
#pragma unroll 2
    for (int j = 0; j < Jn; ++j) {
        const float kv = kp[(size_t)j * Cn];
#pragma unroll
        for (int r = 0; r < IT; ++r) {
            float t = __builtin_fmaf(qs[r], kv, bs);
            float e = __builtin_amdgcn_exp2f(t);        // v_exp_f32
            acc[r] += __builtin_amdgcn_rcpf(1.0f + e);  // v_rcp_f32
        }
    }

#pragma unroll
    for (int r = 0; r < IT; ++r)
        out[((size_t)b * In + (size_t)(i0 + r)) * Cn + c] = acc[r] * (1.0f / Jn);
#endif
}

// ---------------------------------------------------------------------------
extern "C" void kernel_launch(void* const* d_in, const int* in_sizes, int n_in,
                              void* d_out, int out_size, void* d_ws, size_t ws_size,
                              hipStream_t stream) {
    (void)in_sizes; (void)n_in; (void)out_size; (void)d_ws; (void)ws_size;

    const float* Q    = (const float*)d_in[0];   // [B,I,C]
    const float* K    = (const float*)d_in[1];   // [B,J,C]
    const float* bias = (const float*)d_in[2];   // [C]

    float* out  = (float*)d_out;                       // output [B,I,C] first
    float* attn = out + (size_t)Bn * In * Cn;          // then attention_logits [B,I,J]

    // Kernel A: 4 x 32 x 32 wave32 blocks, one 16x16 tile each
    dim3 gA(Jn / 16, In / 16, Bn);
    attn_logits_wmma<<<gA, dim3(32, 1, 1), 0, stream>>>(Q, K, attn);

    // Kernel B: 2 x 64 x 4 blocks of 256 threads
    dim3 gB(Cn / 256, In / IT, Bn);
    sigmoid_mean<<<gB, dim3(256, 1, 1), 0, stream>>>(Q, K, bias, out);
}